// MyNormConv2d_80874234183958
// MI455X (gfx1250) — compile-verified
//
#include <hip/hip_runtime.h>
#include <stdint.h>

#define NB 16
#define NC 3
#define HH 1024
#define WW 1024
#define TH 8                    // output rows per block
#define TILES_Y (HH / TH)

typedef __attribute__((ext_vector_type(4))) unsigned int v4u_t;
typedef __attribute__((ext_vector_type(8))) int          v8i_t;
typedef __attribute__((ext_vector_type(4))) int          v4i_t;

__global__ __launch_bounds__(256)
void normconv3x3_tdm_gfx1250(const float* __restrict__ x,
                             const float* __restrict__ wgt,
                             float* __restrict__ out) {
  // front pad (4) + (TH+2) full rows + tail pad (4): pads keep the t==0 / t==255
  // edge reads in-bounds (their values are masked to zero anyway).
  __shared__ float lds[4 + (TH + 2) * WW + 4];

  const int t  = threadIdx.x;        // 0..255
  const int c0 = t * 4;              // first output column of this thread
  const int ty = blockIdx.x;
  const int ch = blockIdx.y;
  const int n  = blockIdx.z;
  const int y0 = ty * TH;

  const size_t planeOff = (size_t)(n * NC + ch) * (size_t)(HH * WW);
  const float* xplane = x + planeOff;
  float*       oplane = out + planeOff;

  // Clamped input-row range [yStart, yEnd] (tile needs rows y0-1 .. y0+TH).
  int yStart = y0 - 1; if (yStart < 0) yStart = 0;
  int yEnd   = y0 + TH; if (yEnd > HH - 1) yEnd = HH - 1;
  const int rows      = yEnd - yStart + 1;        // TH+1 at borders, else TH+2
  const int ldsRowOff = yStart - (y0 - 1);        // 0 or 1
  float*       ldsBase = &lds[4 + ldsRowOff * WW];
  const float* src     = xplane + (size_t)yStart * WW;

  // Zero-fill halo rows that fall outside the image (reference uses zero pad).
  if (y0 == 0)       *(float4*)&lds[4 + c0]                  = float4{0.f, 0.f, 0.f, 0.f};
  if (y0 + TH == HH) *(float4*)&lds[4 + (TH + 1) * WW + c0]  = float4{0.f, 0.f, 0.f, 0.f};

#if defined(__AMDGCN__) && __has_builtin(__builtin_amdgcn_tensor_load_to_lds) && __has_builtin(__builtin_amdgcn_s_wait_tensorcnt)
  {
    // One wave per block issues the TDM DMA (scalar branch via readfirstlane
    // so waves 1..7 genuinely skip the tensor op), then waits TENSORcnt==0.
    const int wave = __builtin_amdgcn_readfirstlane(t) >> 5;
    if (wave == 0) {
      const unsigned nElems = (unsigned)(rows * WW);     // contiguous 1-D tile
      const unsigned long long ga = (unsigned long long)(uintptr_t)src;
      const unsigned ldsAddr = (unsigned)(uintptr_t)(void*)ldsBase; // low32 = LDS offset

      v4u_t g0;
      g0.x = 1u;                                         // count=1 (valid user D#)
      g0.y = ldsAddr;                                    // lds_addr [63:32]
      g0.z = (unsigned)ga;                               // global_addr [95:64]
      g0.w = (unsigned)((ga >> 32) & 0x01FFFFFFull)      // global_addr[56:32]
           | 0x80000000u;                                // type=2 ("image") [127:126]

      v8i_t g1;
      g1[0] = (int)(2u << 16);                           // data_size=4B; mask=0
      g1[1] = (int)((nElems & 0xFFFFu) << 16);           // tensor_dim0[15:0]
      g1[2] = (int)(((nElems >> 16) & 0xFFFFu)           // tensor_dim0[31:16]
                    | (1u << 16));                       // tensor_dim1 = 1
      g1[3] = (int)((nElems & 0xFFFFu) << 16);           // tile_dim0 = nElems (<=10240)
      g1[4] = 1;                                         // tile_dim1=1, tile_dim2=0
      g1[5] = (int)nElems;                               // tensor_dim0_stride (benign)
      g1[6] = 0;
      g1[7] = 0;

      v4i_t g2; g2.x = 1; g2.y = 1; g2.z = 0; g2.w = 0;  // tensor_dim2/3 = 1 (benign)
      v4i_t g3; g3.x = 0; g3.y = (int)(1u << 16); g3.z = 0; g3.w = 0; // tensor_dim4=1

      v8i_t gx; gx[0]=0; gx[1]=0; gx[2]=0; gx[3]=0; gx[4]=0; gx[5]=0; gx[6]=0; gx[7]=0;

      // amdgpu-toolchain (clang-23) form: 6 args, trailing i32 = cache policy.
      __builtin_amdgcn_tensor_load_to_lds(g0, g1, g2, g3, gx, 0);
      __builtin_amdgcn_s_wait_tensorcnt(0);
    }
  }
#else
  // Fallback: cooperative contiguous b128 copy (tile rows are contiguous).
  for (int i = c0; i < rows * WW; i += 256 * 4)
    *(float4*)(ldsBase + i) = *(const float4*)(src + i);
#endif
  __syncthreads();

  // Normalized weights for this channel (wave-uniform scalar loads).
  float wv[9];
  float s2 = 0.f;
#pragma unroll
  for (int k = 0; k < 9; ++k) { float w = wgt[ch * 9 + k]; wv[k] = w; s2 += w * w; }
  const float wn = sqrtf(s2);
#pragma unroll
  for (int k = 0; k < 9; ++k) wv[k] /= wn;

  // 3-phase rolling window: v[p][0..5] = row values, h[p][0..3] = horizontal
  // sum-of-squares. Full unroll turns the p-rotation into register renaming.
  float v[3][6];
  float h[3][4];

  auto loadRow = [&](int rr, float (&vr)[6], float (&hr)[4]) {
    const float* rb = &lds[4 + rr * WW];
    float4 B    = *(const float4*)(rb + c0);    // cols c0..c0+3 (16B aligned)
    float left  = rb[c0 - 1];                   // col c0-1 (pad keeps in-bounds)
    float right = rb[c0 + 4];                   // col c0+4
    if (t == 0)   left  = 0.f;                  // col -1   -> zero pad
    if (t == 255) right = 0.f;                  // col 1024 -> zero pad
    vr[0] = left; vr[1] = B.x; vr[2] = B.y; vr[3] = B.z; vr[4] = B.w; vr[5] = right;
    float q0 = vr[0]*vr[0], q1 = vr[1]*vr[1], q2 = vr[2]*vr[2];
    float q3 = vr[3]*vr[3], q4 = vr[4]*vr[4], q5 = vr[5]*vr[5];
    float t12 = q1 + q2, t34 = q3 + q4;
    hr[0] = q0 + t12; hr[1] = t12 + q3;
    hr[2] = q2 + t34; hr[3] = t34 + q5;
  };

  loadRow(0, v[0], h[0]);
  loadRow(1, v[1], h[1]);

#pragma unroll
  for (int j = 0; j < TH; ++j) {
    const int ia = j % 3;            // top row    (y-1)
    const int ib = (j + 1) % 3;      // middle row (y)
    const int ic = (j + 2) % 3;      // bottom row (y+1)
    loadRow(j + 2, v[ic], h[ic]);

    const float (&va)[6] = v[ia]; const float (&ha)[4] = h[ia];
    const float (&vb)[6] = v[ib]; const float (&hb)[4] = h[ib];
    const float (&vc)[6] = v[ic]; const float (&hc)[4] = h[ic];

    float r[4];
#pragma unroll
    for (int k = 0; k < 4; ++k) {
      float num = wv[0]*va[k] + wv[1]*va[k+1] + wv[2]*va[k+2]
                + wv[3]*vb[k] + wv[4]*vb[k+1] + wv[5]*vb[k+2]
                + wv[6]*vc[k] + wv[7]*vc[k+1] + wv[8]*vc[k+2];
      float sq = ha[k] + hb[k] + hc[k];
#if defined(__AMDGCN__) && __has_builtin(__builtin_amdgcn_rsqf)
      r[k] = num * __builtin_amdgcn_rsqf(sq);   // v_rsq_f32, ~1 ulp
#else
      r[k] = num / sqrtf(sq);
#endif
    }
    *(float4*)(oplane + (size_t)(y0 + j) * WW + c0) = float4{r[0], r[1], r[2], r[3]};
  }
}

extern "C" void kernel_launch(void* const* d_in, const int* in_sizes, int n_in,
                              void* d_out, int out_size, void* d_ws, size_t ws_size,
                              hipStream_t stream) {
  const float* x   = (const float*)d_in[0];   // (16,3,1024,1024) f32
  const float* wgt = (const float*)d_in[1];   // (3,1,3,3) f32
  float*       out = (float*)d_out;           // (16,3,1024,1024) f32
  (void)in_sizes; (void)n_in; (void)out_size; (void)d_ws; (void)ws_size;

  dim3 grid(TILES_Y, NC, NB);   // 128 x 3 x 16 = 6144 blocks
  dim3 block(256, 1, 1);        // 8 waves (wave32)
  normconv3x3_tdm_gfx1250<<<grid, block, 0, stream>>>(x, wgt, out);
}